// WeisfeilerLehman_56573309223907
// MI455X (gfx1250) — compile-verified
//
#include <hip/hip_runtime.h>
#include <stdint.h>

typedef unsigned int u32;
typedef unsigned short u16;
typedef unsigned long long u64;

typedef u32 tdm_v4u __attribute__((ext_vector_type(4)));
typedef int tdm_v8i __attribute__((ext_vector_type(8)));
typedef int tdm_v4i __attribute__((ext_vector_type(4)));

#define DCLS 128
#define TILE_ROWS 64
#define POW_TAB 4096

// ---------------------------------------------------------------------------
// TDM: issue one 2D tile load (rows x 128 f32, row-major, contiguous) to LDS.
// D# group0: count=1 | lds_addr | global_addr(57b) | type=2
// D# group1: data_size=4B, tensor_dim0=128, tensor_dim1=N, tile_dim0=128,
//            tile_dim1=rows, tensor_dim0_stride=128.  groups 2/3 = 0 (2D).
// clang-23 builtin arity: (v4u, v8i, v4i, v4i, v8i, i32 cpol)
// ---------------------------------------------------------------------------
__device__ __forceinline__ void tdm_issue_tile(u32 lds_off, const float* gsrc,
                                               int rows, int Ntotal) {
  const u64 ga = (u64)(uintptr_t)gsrc;
  const u32 td0 = (u32)DCLS;         // tensor dim0 (elements)
  const u32 td1 = (u32)Ntotal;       // tensor dim1 (rows)
  tdm_v4u g0 = {
    1u,                                                // count=1, user mode
    lds_off,                                           // lds_addr (bytes)
    (u32)ga,                                           // global_addr[31:0]
    (u32)((ga >> 32) & 0x1FFFFFFull) | (2u << 30)      // ga[56:32] | type=2
  };
  tdm_v8i g1 = {
    (int)(2u << 16),                                          // data_size=4B
    (int)((td0 & 0xFFFFu) << 16),                             // tensor_dim0 lo
    (int)(((td0 >> 16) & 0xFFFFu) | ((td1 & 0xFFFFu) << 16)), // td0 hi | td1 lo
    (int)(((td1 >> 16) & 0xFFFFu) | ((u32)DCLS << 16)),       // td1 hi | tile_dim0
    (int)((u32)rows & 0xFFFFu),                               // tile_dim1 | tile_dim2=0
    (int)DCLS,                                                // tensor_dim0_stride lo32
    0, 0
  };
  tdm_v4i z4 = {0, 0, 0, 0};
  tdm_v8i z8 = {0, 0, 0, 0, 0, 0, 0, 0};
  __builtin_amdgcn_tensor_load_to_lds(g0, g1, z4, z4, z8, 0);
}

// ---------------------------------------------------------------------------
// Argmax over D=128 per node, TDM double-buffered through LDS.
// One wave per row: 32 lanes x float4 (ds_load_b128) + shfl_xor butterfly.
// Writes labels[] (u64) and history slot h0 (floats at out_h0).
// ---------------------------------------------------------------------------
__global__ __launch_bounds__(256) void k_argmax_tdm(const float* __restrict__ x,
                                                    u64* __restrict__ labels,
                                                    float* __restrict__ out_h0,
                                                    int N) {
  __shared__ float buf[2][TILE_ROWS * DCLS];   // 2 x 32KB
  const int ntiles = (N + TILE_ROWS - 1) / TILE_ROWS;
  const int tid = threadIdx.x;
  int t = (int)blockIdx.x;
  if (t >= ntiles) return;

  if (tid < 32) {   // wave 0 drives the TDM
    const int rows0 = (N - t * TILE_ROWS) < TILE_ROWS ? (N - t * TILE_ROWS) : TILE_ROWS;
    tdm_issue_tile((u32)(uintptr_t)&buf[0][0], x + (size_t)t * TILE_ROWS * DCLS, rows0, N);
  }

  int parity = 0;
  for (; t < ntiles; t += (int)gridDim.x) {
    const int tn = t + (int)gridDim.x;
    if (tid < 32) {
      if (tn < ntiles) {
        const int rowsn = (N - tn * TILE_ROWS) < TILE_ROWS ? (N - tn * TILE_ROWS) : TILE_ROWS;
        tdm_issue_tile((u32)(uintptr_t)&buf[parity ^ 1][0],
                       x + (size_t)tn * TILE_ROWS * DCLS, rowsn, N);
        __builtin_amdgcn_s_wait_tensorcnt(1);   // current tile done, next in flight
      } else {
        __builtin_amdgcn_s_wait_tensorcnt(0);
      }
    }
    asm volatile("" ::: "memory");  // TDM wrote LDS behind the compiler's back
    __syncthreads();

    const int wave = tid >> 5;
    const int lane = tid & 31;
    for (int rloc = wave; rloc < TILE_ROWS; rloc += 8) {
      const float4 q = *reinterpret_cast<const float4*>(&buf[parity][rloc * DCLS + lane * 4]);
      float v = q.x; int idx = lane * 4;
      if (q.y > v) { v = q.y; idx = lane * 4 + 1; }
      if (q.z > v) { v = q.z; idx = lane * 4 + 2; }
      if (q.w > v) { v = q.w; idx = lane * 4 + 3; }
      // wave32 butterfly argmax, first-occurrence tie-break
      for (int m = 16; m; m >>= 1) {
        const float ov = __shfl_xor(v, m, 32);
        const int   oi = __shfl_xor(idx, m, 32);
        if (ov > v || (ov == v && oi < idx)) { v = ov; idx = oi; }
      }
      const int gnode = t * TILE_ROWS + rloc;
      if (lane == 0 && gnode < N) {
        labels[gnode] = (u64)idx;
        out_h0[gnode] = (float)idx;
      }
    }
    __syncthreads();
    parity ^= 1;
  }
}

// ---------------------------------------------------------------------------
// Setup kernels for the ordered per-edge exponents
// ---------------------------------------------------------------------------
__global__ void k_zero_u32(u32* __restrict__ p, int n) {
  const int i = blockIdx.x * blockDim.x + threadIdx.x;
  if (i < n) p[i] = 0u;
}

__global__ void k_pow_table(u64* __restrict__ tab) {
  const int i = blockIdx.x * blockDim.x + threadIdx.x;
  if (i < POW_TAB) {
    u32 ex = (u32)i;
    u64 b = 31ull, r = 1ull;
    while (ex) { if (ex & 1u) r *= b; b *= b; ex >>= 1u; }
    tab[i] = r;                       // 31^i mod 2^64
  }
}

__global__ void k_deg(const int* __restrict__ row, u32* __restrict__ deg, int E) {
  const int e = blockIdx.x * blockDim.x + threadIdx.x;
  if (e < E) atomicAdd(&deg[row[e]], 1u);
}

__global__ void k_alloc(const u32* __restrict__ deg, u32* __restrict__ start,
                        u32* __restrict__ cursor, int N) {
  const int r = blockIdx.x * blockDim.x + threadIdx.x;
  if (r < N) start[r] = atomicAdd(cursor, deg[r]);   // contiguous, placement-order-free
}

__global__ void k_scatter(const int* __restrict__ row, const u32* __restrict__ start,
                          u32* __restrict__ fill, u32* __restrict__ tmp, int E) {
  const int e = blockIdx.x * blockDim.x + threadIdx.x;
  if (e < E) {
    const int r = row[e];
    const u32 slot = atomicAdd(&fill[r], 1u);
    tmp[start[r] + slot] = (u32)e;
  }
}

// One wave per row: rank(edge) = #smaller edge-ids in its segment (stable order),
// expo[e] = deg-1-rank  (u16, looked up in the 31^k table each iteration).
__global__ void k_rank(const u32* __restrict__ start, const u32* __restrict__ deg,
                       const u32* __restrict__ tmp, u16* __restrict__ expo, int N) {
  const int lane = threadIdx.x & 31;
  const int wid = (int)((blockIdx.x * blockDim.x + threadIdx.x) >> 5);
  if (wid >= N) return;
  const u32 s = start[wid];
  const u32 d = deg[wid];
  for (u32 i = (u32)lane; i < d; i += 32u) {
    const u32 eid = tmp[s + i];
    u32 cnt = 0;
    for (u32 j = 0; j < d; ++j) cnt += (tmp[s + j] < eid) ? 1u : 0u;
    expo[eid] = (u16)(d - 1u - cnt);
  }
}

// ---------------------------------------------------------------------------
// Per-iteration: ordered-hash segment sum (exact mod 2^64 via u64 atomics)
// ---------------------------------------------------------------------------
__global__ void k_edge_accum(const int* __restrict__ row, const int* __restrict__ col,
                             const u16* __restrict__ expo, const u64* __restrict__ tab,
                             const u64* __restrict__ labels, u64* __restrict__ newl,
                             int E) {
  const int e = blockIdx.x * blockDim.x + threadIdx.x;
  if (e < E) {
    const u64 v = labels[col[e]] * tab[expo[e]]; // wraps mod 2^64 == torch int64
    atomicAdd(&newl[row[e]], v);                 // global_atomic_add_u64 (L2-resident)
  }
}

// Commit labels and re-zero newl for the next iteration (fused zeroing).
__global__ void k_commit(u64* __restrict__ newl, u64* __restrict__ labels,
                         float* __restrict__ out, int N, int histOff, int alsoFinal) {
  const int n = blockIdx.x * blockDim.x + threadIdx.x;
  if (n < N) {
    const u64 v = newl[n];
    newl[n] = 0ull;                        // ready for next iteration / next replay
    labels[n] = v;
    const float f = (float)(long long)v;   // harness output buffer is float
    out[histOff + n] = f;
    if (alsoFinal) out[n] = f;
  }
}

// ---------------------------------------------------------------------------
extern "C" void kernel_launch(void* const* d_in, const int* in_sizes, int n_in,
                              void* d_out, int out_size, void* d_ws, size_t ws_size,
                              hipStream_t stream) {
  const float* x  = (const float*)d_in[0];
  const int*   ei = (const int*)d_in[1];          // edge_index [2, E]
  const int N = in_sizes[0] / DCLS;
  const int E = in_sizes[1] / 2;
  const int* row = ei;
  const int* col = ei + E;
  float* out = (float*)d_out;   // layout: [final | h0 | h1 | h2 | h3], N each

  // workspace carve: newl|deg|fill|cursor contiguous so one kernel zeroes all
  char* w = (char*)d_ws;
  u64* labels = (u64*)w;  w += (size_t)N * 8;
  u64* newl   = (u64*)w;  w += (size_t)N * 8;
  u32* deg    = (u32*)w;  w += (size_t)N * 4;
  u32* fill   = (u32*)w;  w += (size_t)N * 4;
  u32* cursor = (u32*)w;  w += 16;
  u32* start  = (u32*)w;  w += (size_t)N * 4;
  u32* tmp    = (u32*)w;  w += (size_t)E * 4;
  u16* expo   = (u16*)w;  w += (size_t)E * 2;
  u64* tab    = (u64*)w;                          // POW_TAB * 8

  const int TPB = 256;
  auto cdiv = [](int a, int b) { return (a + b - 1) / b; };

  // --- one-time setup ---
  k_zero_u32 <<<cdiv(4 * N + 4, TPB), TPB, 0, stream>>>(
      (u32*)newl, 4 * N + 4);                           // newl + deg + fill + cursor
  k_pow_table<<<cdiv(POW_TAB, TPB), TPB, 0, stream>>>(tab);
  k_deg      <<<cdiv(E, TPB), TPB, 0, stream>>>(row, deg, E);
  k_alloc    <<<cdiv(N, TPB), TPB, 0, stream>>>(deg, start, cursor, N);
  k_scatter  <<<cdiv(E, TPB), TPB, 0, stream>>>(row, start, fill, tmp, E);
  k_rank     <<<cdiv(N, 8), TPB, 0, stream>>>(start, deg, tmp, expo, N); // 8 waves/blk

  // --- argmax labels via TDM-staged tiles (writes labels + history[0]) ---
  const int ntiles = cdiv(N, TILE_ROWS);
  const int nblk = ntiles < 512 ? ntiles : 512;
  k_argmax_tdm<<<nblk, 256, 0, stream>>>(x, labels, out + N, N);

  // --- 3 WL iterations ---
  for (int it = 0; it < 3; ++it) {
    k_edge_accum<<<cdiv(E, TPB), TPB, 0, stream>>>(row, col, expo, tab, labels, newl, E);
    k_commit    <<<cdiv(N, TPB), TPB, 0, stream>>>(newl, labels, out, N,
                                                   (2 + it) * N, it == 2 ? 1 : 0);
  }
}